// ImputationModule_89524298318174
// MI455X (gfx1250) — compile-verified
//
#include <hip/hip_runtime.h>

// Problem constants from the reference: D=1024, K=8, N = rows (8192 in harness).
#define DD 1024
#define KK 8
#define ROWS_PER_BLOCK 8
#define NTHREADS 256   // 8 waves of 32; each thread owns 4 consecutive columns -> 256*4 = D

// ---- CDNA5 async global->LDS copy (ASYNCcnt-tracked), 128-bit per lane ----
// VDST VGPR = per-lane LDS byte address; VADDR = 64-bit global address pair.
// Flat pointers into LDS carry the LDS byte offset in their low 32 bits
// (ISA 10.2), so truncation gives the LDS address.
__device__ __forceinline__ void async_copy_b128(const float* gsrc, float* ldst) {
  unsigned lds_addr = (unsigned)(unsigned long long)ldst;
  asm volatile("global_load_async_to_lds_b128 %0, %1, off"
               :: "v"(lds_addr), "v"(gsrc)
               : "memory");
}

template <int N>
__device__ __forceinline__ void wait_async() {
#if __has_builtin(__builtin_amdgcn_s_wait_asynccnt)
  __builtin_amdgcn_s_wait_asynccnt((unsigned short)N);
  asm volatile("" ::: "memory");   // keep LDS reads below the wait
#else
  asm volatile("s_wait_asynccnt %0" :: "i"(N) : "memory");
#endif
}

// ---- Kernel 1: pull the block-diagonal of W into workspace: wd[m*D + i] = W[i, m*D + i]
__global__ void wdiag_extract_kernel(const float* __restrict__ W, float* __restrict__ wd) {
  int j = blockIdx.x * blockDim.x + threadIdx.x;   // 0 .. K*D-1
  int m = j >> 10;          // j / D
  int i = j & (DD - 1);     // j % D
  wd[j] = W[(size_t)i * (KK * DD) + (size_t)m * DD + i];
}

// ---- Kernel 2: streaming diagonal-weighted reduction with async double-buffering
__global__ __launch_bounds__(NTHREADS) void imput_rows_kernel(
    const float* __restrict__ h, const float* __restrict__ wd,
    const float* __restrict__ b, float* __restrict__ out, int nrows) {
  // lds[buf][wave][m][lane][4] : 2 * 8 * 8 * 32 * 4 floats = 64 KB
  __shared__ float lds[2 * 8 * KK * 32 * 4];

  const int t    = threadIdx.x;
  const int wave = t >> 5;
  const int lane = t & 31;
  const int i0   = t << 2;                       // column base (16B aligned)

  // Per-thread diagonal weights + bias live in registers for the whole block.
  float4 w[KK];
#pragma unroll
  for (int m = 0; m < KK; ++m)
    w[m] = *(const float4*)(wd + m * DD + i0);
  const float4 bias = *(const float4*)(b + i0);

  const long row0    = (long)blockIdx.x * ROWS_PER_BLOCK;
  const float* hbase = h + row0 * (KK * DD) + i0;
  float* const wbase = lds + wave * (KK * 32 * 4);   // this wave's private region
  const int llane = lane << 2;

  if (row0 + ROWS_PER_BLOCK <= nrows) {
    // Stage row r into buffer (8 async b128 per wave = this wave's 4KB slice).
    auto stage = [&](int r, int buf) {
#pragma unroll
      for (int m = 0; m < KK; ++m)
        async_copy_b128(hbase + (long)r * (KK * DD) + m * DD,
                        wbase + buf * (8 * KK * 32 * 4) + m * 128 + llane);
    };

    stage(0, 0);
#pragma unroll
    for (int r = 0; r < ROWS_PER_BLOCK; ++r) {
      const int buf = r & 1;
      if (r + 1 < ROWS_PER_BLOCK) {
        stage(r + 1, buf ^ 1);
        wait_async<KK>();   // drain row r's 8 loads; keep row r+1's 8 in flight
      } else {
        wait_async<0>();
      }
      float4 acc = bias;
#pragma unroll
      for (int m = 0; m < KK; ++m) {
        const float4 v =
            *(const float4*)(wbase + buf * (8 * KK * 32 * 4) + m * 128 + llane);
        acc.x = fmaf(v.x, w[m].x, acc.x);
        acc.y = fmaf(v.y, w[m].y, acc.y);
        acc.z = fmaf(v.z, w[m].z, acc.z);
        acc.w = fmaf(v.w, w[m].w, acc.w);
      }
      *(float4*)(out + (row0 + r) * DD + i0) = acc;
    }
  } else {
    // Tail (not hit for N=8192): plain coalesced path with bounds check.
    for (int r = 0; row0 + r < nrows; ++r) {
      const float* hp = hbase + (long)r * (KK * DD);
      float4 acc = bias;
#pragma unroll
      for (int m = 0; m < KK; ++m) {
        const float4 v = *(const float4*)(hp + m * DD);
        acc.x = fmaf(v.x, w[m].x, acc.x);
        acc.y = fmaf(v.y, w[m].y, acc.y);
        acc.z = fmaf(v.z, w[m].z, acc.z);
        acc.w = fmaf(v.w, w[m].w, acc.w);
      }
      *(float4*)(out + (row0 + r) * DD + i0) = acc;
    }
  }
}

extern "C" void kernel_launch(void* const* d_in, const int* in_sizes, int n_in,
                              void* d_out, int out_size, void* d_ws, size_t ws_size,
                              hipStream_t stream) {
  const float* h_r = (const float*)d_in[0];   // (N, K*D) fp32
  const float* W   = (const float*)d_in[1];   // (D, K*D) fp32
  const float* b   = (const float*)d_in[2];   // (D,)    fp32
  float* out = (float*)d_out;                 // (N, D)  fp32
  float* wd  = (float*)d_ws;                  // 32 KB scratch: w_diag[K][D]

  const int N = in_sizes[0] / (KK * DD);

  wdiag_extract_kernel<<<dim3((KK * DD) / 256), dim3(256), 0, stream>>>(W, wd);

  const int nblocks = (N + ROWS_PER_BLOCK - 1) / ROWS_PER_BLOCK;
  imput_rows_kernel<<<dim3(nblocks), dim3(NTHREADS), 0, stream>>>(h_r, wd, b, out, N);
}